// GCNLayer_31688268710208
// MI455X (gfx1250) — compile-verified
//
#include <hip/hip_runtime.h>

#define D_FEAT 64
#define EPB 256   // edges per block = 8 waves * 32 lanes

__global__ void gcn_zero_out_kernel(float4* __restrict__ out4, int n4) {
  int i = blockIdx.x * blockDim.x + threadIdx.x;
  int stride = gridDim.x * blockDim.x;
  const float4 z = make_float4(0.f, 0.f, 0.f, 0.f);
  for (; i < n4; i += stride) out4[i] = z;
}

__global__ void __launch_bounds__(256)
gcn_spmm_scatter_kernel(const int* __restrict__ rows,
                        const int* __restrict__ cols,
                        const float* __restrict__ vals,
                        const float* __restrict__ embeds,
                        float* __restrict__ out,
                        int nEdges) {
  __shared__ int   s_row[EPB];
  __shared__ int   s_col[EPB];
  __shared__ float s_val[EPB];

  const int tid       = threadIdx.x;
  const int lane      = tid & 31;
  const int wave      = tid >> 5;
  const int blockBase = blockIdx.x * EPB;

  // ---- Stage this wave's 32 edge triples into its private LDS slice using
  //      CDNA5 async global->LDS copies (ASYNCcnt), one b32 per lane per array.
  {
    int e  = blockBase + wave * 32 + lane;
    int ec = e < nEdges ? e : (nEdges - 1);      // clamp tail to a valid address
    unsigned voff = (unsigned)ec * 4u;           // byte offset into edge arrays
    unsigned ldsR = (unsigned)(size_t)&s_row[wave * 32 + lane];
    unsigned ldsC = (unsigned)(size_t)&s_col[wave * 32 + lane];
    unsigned ldsV = (unsigned)(size_t)&s_val[wave * 32 + lane];
    asm volatile("global_load_async_to_lds_b32 %0, %1, %2"
                 :: "v"(ldsR), "v"(voff), "s"(rows) : "memory");
    asm volatile("global_load_async_to_lds_b32 %0, %1, %2"
                 :: "v"(ldsC), "v"(voff), "s"(cols) : "memory");
    asm volatile("global_load_async_to_lds_b32 %0, %1, %2"
                 :: "v"(ldsV), "v"(voff), "s"(vals) : "memory");
    asm volatile("s_wait_asynccnt 0" ::: "memory");
    // Wave-private LDS region: issuing wave is the only reader, no barrier needed.
  }

  const int half = lane >> 4;        // two edges in flight per wave iteration
  const int f    = (lane & 15) * 4;  // 16 lanes x float4 = 64 coalesced floats/edge

  if (blockBase + EPB <= nEdges) {
    // -------- Fast path: full block, no per-edge bounds checks. --------
    #pragma unroll 4
    for (int i = 0; i < 16; ++i) {
      const int e = wave * 32 + i * 2 + half;
      const int c = s_col[e];
      const int r = s_row[e];
      const float v = s_val[e];

      // Warm L2/L0 for a future edge's embedding row (global_prefetch_b8).
      const int epf = wave * 32 + ((i * 2 + half + 8) & 31);
      __builtin_prefetch(embeds + (size_t)s_col[epf] * D_FEAT + f, 0, 1);

      const float4 x = *reinterpret_cast<const float4*>(
          embeds + (size_t)c * D_FEAT + f);
      float* o = out + (size_t)r * D_FEAT + f;

      // Hardware f32 atomic adds; resolve in the 192MB L2 (output is resident).
      unsafeAtomicAdd(o + 0, x.x * v);
      unsafeAtomicAdd(o + 1, x.y * v);
      unsafeAtomicAdd(o + 2, x.z * v);
      unsafeAtomicAdd(o + 3, x.w * v);
    }
  } else {
    // -------- Tail path: guard every edge (only the last block). --------
    for (int i = 0; i < 16; ++i) {
      const int e = wave * 32 + i * 2 + half;
      const int g = blockBase + e;
      if (g < nEdges) {
        const int c = s_col[e];
        const int r = s_row[e];
        const float v = s_val[e];
        const float4 x = *reinterpret_cast<const float4*>(
            embeds + (size_t)c * D_FEAT + f);
        float* o = out + (size_t)r * D_FEAT + f;
        unsafeAtomicAdd(o + 0, x.x * v);
        unsafeAtomicAdd(o + 1, x.y * v);
        unsafeAtomicAdd(o + 2, x.z * v);
        unsafeAtomicAdd(o + 3, x.w * v);
      }
    }
  }
}

extern "C" void kernel_launch(void* const* d_in, const int* in_sizes, int n_in,
                              void* d_out, int out_size, void* d_ws, size_t ws_size,
                              hipStream_t stream) {
  const int*   rows   = (const int*)d_in[0];
  const int*   cols   = (const int*)d_in[1];
  const float* vals   = (const float*)d_in[2];
  const float* embeds = (const float*)d_in[3];
  float*       out    = (float*)d_out;

  const int nEdges = in_sizes[0];

  // out_size = N_NODES * 64, divisible by 4.
  const int n4 = out_size / 4;
  const int zBlocks = (n4 + 255) / 256;
  gcn_zero_out_kernel<<<zBlocks, 256, 0, stream>>>((float4*)out, n4);

  const int sBlocks = (nEdges + EPB - 1) / EPB;
  gcn_spmm_scatter_kernel<<<sBlocks, 256, 0, stream>>>(rows, cols, vals, embeds,
                                                       out, nEdges);
}